// MultiTaskLoss_67276367724950
// MI455X (gfx1250) — compile-verified
//
#include <hip/hip_runtime.h>
#include <stdint.h>

#define NB 1024
#define NT 256

#if defined(__HIP_DEVICE_COMPILE__) && \
    __has_builtin(__builtin_amdgcn_global_load_async_to_lds_b32) && \
    __has_builtin(__builtin_amdgcn_s_wait_asynccnt)
#define USE_ASYNC_LDS 1
#else
#define USE_ASYNC_LDS 0
#endif

#if USE_ASYNC_LDS && __has_builtin(__builtin_amdgcn_global_load_async_to_lds_b128)
#define USE_ASYNC_B128 1
#else
#define USE_ASYNC_B128 0
#endif

typedef int v4i __attribute__((ext_vector_type(4)));
typedef __attribute__((address_space(1))) int gint_t;
typedef __attribute__((address_space(3))) int lint_t;
typedef __attribute__((address_space(1))) v4i gv4i_t;
typedef __attribute__((address_space(3))) v4i lv4i_t;

// outstanding async ops per wave per tile
#if USE_ASYNC_B128
#define ASYNC_OPS 4
#else
#define ASYNC_OPS 13
#endif

__device__ __forceinline__ float wave_sum(float v) {
    v += __shfl_xor(v, 16, 32);
    v += __shfl_xor(v, 8, 32);
    v += __shfl_xor(v, 4, 32);
    v += __shfl_xor(v, 2, 32);
    v += __shfl_xor(v, 1, 32);
    return v;
}

// 2-class weighted CE; returns loss and logp(class0)
__device__ __forceinline__ void ce2(float a, float b, int y, float wpos,
                                    float& loss, float& lp0) {
    float m   = fmaxf(a, b);
    float lse = m + __logf(__expf(a - m) + __expf(b - m));
    lp0       = a - lse;
    float ly  = (y != 0 ? b : a) - lse;
    float w   = (y != 0 ? wpos : 1.0f);
    loss      = -w * ly;
}

// 3-class CE with weights {1,2,4}; returns loss and logp(class0)
__device__ __forceinline__ void ce3(float a, float b, float c, int y,
                                    float& loss, float& lp0) {
    float m   = fmaxf(fmaxf(a, b), c);
    float lse = m + __logf(__expf(a - m) + __expf(b - m) + __expf(c - m));
    lp0       = a - lse;
    float ly  = (y == 0 ? a : (y == 1 ? b : c)) - lse;
    float w   = (y == 0 ? 1.0f : (y == 1 ? 2.0f : 4.0f));
    loss      = -w * ly;
}

extern "C" __global__ void __launch_bounds__(NT)
mtl_partial(const float* __restrict__ logits,
            const int*   __restrict__ lbin,
            const int*   __restrict__ ltri,
            float*       __restrict__ partial,
            int nrows)
{
#if USE_ASYNC_LDS
    // double-buffered stage: per wave a contiguous 32 rows * 13 dwords = 416 dwords
    __shared__ __align__(16) float stage[2][NT * 13];
#endif
    __shared__ float red[NT / 32][6];

    const int  tid    = threadIdx.x;
    const int  lane   = tid & 31;
    const int  wid    = tid >> 5;
    long long  row    = (long long)blockIdx.x * NT + tid;
    const long long stride = (long long)gridDim.x * NT;

#if USE_ASYNC_LDS
    // Coalesced async fill of this wave's 416-dword region (identical linear
    // offsets in global and LDS). Each wave reads back only its own region,
    // so only s_wait_asynccnt (per-wave counter) is needed — no block barrier.
    auto issue = [&](long long r, int b) {
        long long wrow = r - lane;                 // first row owned by this wave
        const float* g = logits + wrow * 13;
        float*       l = &stage[b][wid * (32 * 13)];
#if USE_ASYNC_B128
        // 104 16B chunks: 3 full-wave b128 ops + 1 op with lanes 0..7 active.
        // ASYNCcnt is per-wave -> uniformly ASYNC_OPS=4 outstanding per tile.
#pragma unroll
        for (int it = 0; it < 3; ++it) {
            int o = (it * 32 + lane) * 4;
            __builtin_amdgcn_global_load_async_to_lds_b128(
                (gv4i_t*)(g + o), (lv4i_t*)(l + o), 0, 0);
        }
        {
            int o = (96 + lane) * 4;
            if (lane < 8)
                __builtin_amdgcn_global_load_async_to_lds_b128(
                    (gv4i_t*)(g + o), (lv4i_t*)(l + o), 0, 0);
        }
#else
#pragma unroll
        for (int it = 0; it < 13; ++it) {
            int o = it * 32 + lane;
            __builtin_amdgcn_global_load_async_to_lds_b32(
                (gint_t*)(g + o), (lint_t*)(l + o), 0, 0);
        }
#endif
    };
#endif

    float a0 = 0.f, a1 = 0.f, a2 = 0.f, a3 = 0.f, a4 = 0.f, a5 = 0.f;
    bool have = row < nrows;
    int  buf  = 0;
#if USE_ASYNC_LDS
    if (have) issue(row, 0);
#endif
    while (have) {
        long long nrow  = row + stride;
        bool      hnext = nrow < (long long)nrows;
#if USE_ASYNC_LDS
        if (hnext) {
            // make sure prior LDS reads of buf^1 retired before DMA overwrites it
            asm volatile("s_wait_dscnt 0" ::: "memory");
            issue(nrow, buf ^ 1);
        }
#endif
        // labels: independent global stream, overlaps with async wait
        int y0 = __builtin_nontemporal_load(&lbin[row * 2 + 0]);
        int y1 = __builtin_nontemporal_load(&lbin[row * 2 + 1]);
        int y2 = __builtin_nontemporal_load(&ltri[row * 3 + 0]);
        int y3 = __builtin_nontemporal_load(&ltri[row * 3 + 1]);
        int y4 = __builtin_nontemporal_load(&ltri[row * 3 + 2]);

#if USE_ASYNC_LDS
        if (hnext) __builtin_amdgcn_s_wait_asynccnt(ASYNC_OPS); // next tile stays in flight
        else       __builtin_amdgcn_s_wait_asynccnt(0);
        const float* L = &stage[buf][tid * 13];          // stride 13 dwords: conflict-free
#else
        const float* L = logits + row * 13;
#endif
        float l, p, minlp;
        ce2(L[0],  L[1],        y0, 2.0f, l, p); a0 += l; minlp = p;
        ce2(L[2],  L[3],        y1, 6.0f, l, p); a1 += l; minlp = fminf(minlp, p);
        ce3(L[4],  L[5],  L[6], y2,       l, p); a2 += l; minlp = fminf(minlp, p);
        ce3(L[7],  L[8],  L[9], y3,       l, p); a3 += l; minlp = fminf(minlp, p);
        ce3(L[10], L[11], L[12], y4,      l, p); a4 += l; minlp = fminf(minlp, p);

        // p = max_i(1-exp(lp0_i))  =>  log1p(-p) = min_i lp0_i ; log(p) = log1p(-exp(minlp))
        int anyy = (y0 | y1 | y2 | y3 | y4);
        a5 += (anyy != 0) ? (-6.0f * log1pf(-__expf(minlp))) : (-minlp);

        row = nrow; have = hnext; buf ^= 1;
    }

    float acc[6] = {a0, a1, a2, a3, a4, a5};
#pragma unroll
    for (int j = 0; j < 6; ++j) acc[j] = wave_sum(acc[j]);
    if (lane == 0) {
#pragma unroll
        for (int j = 0; j < 6; ++j) red[wid][j] = acc[j];
    }
    __syncthreads();
    if (tid < 6) {
        float s = 0.f;
#pragma unroll
        for (int w = 0; w < NT / 32; ++w) s += red[w][tid];
        partial[(long long)blockIdx.x * 6 + tid] = s;
    }
}

extern "C" __global__ void __launch_bounds__(NT)
mtl_final(const float* __restrict__ partial, float* __restrict__ out,
          int nblocks, float invB)
{
    __shared__ float red[NT / 32][6];
    float acc[6] = {0.f, 0.f, 0.f, 0.f, 0.f, 0.f};
    for (int b = threadIdx.x; b < nblocks; b += NT) {
#pragma unroll
        for (int j = 0; j < 6; ++j) acc[j] += partial[(long long)b * 6 + j];
    }
#pragma unroll
    for (int j = 0; j < 6; ++j) acc[j] = wave_sum(acc[j]);
    int wid = threadIdx.x >> 5, lane = threadIdx.x & 31;
    if (lane == 0) {
#pragma unroll
        for (int j = 0; j < 6; ++j) red[wid][j] = acc[j];
    }
    __syncthreads();
    if (threadIdx.x == 0) {
        float tot = 0.f;
#pragma unroll
        for (int j = 0; j < 6; ++j) {
            float s = 0.f;
#pragma unroll
            for (int w = 0; w < NT / 32; ++w) s += red[w][j];
            s *= invB;
            out[j] = s;
            tot += s;
        }
        out[6] = tot;
    }
}

extern "C" void kernel_launch(void* const* d_in, const int* in_sizes, int n_in,
                              void* d_out, int out_size, void* d_ws, size_t ws_size,
                              hipStream_t stream) {
    const float* logits = (const float*)d_in[0];
    const int*   lbin   = (const int*)d_in[1];
    const int*   ltri   = (const int*)d_in[2];
    float*       out    = (float*)d_out;
    float*       part   = (float*)d_ws;   // NB*6 floats = 24 KB scratch
    int nrows = in_sizes[0] / 13;

    mtl_partial<<<NB, NT, 0, stream>>>(logits, lbin, ltri, part, nrows);
    mtl_final<<<1, NT, 0, stream>>>(part, out, NB, 1.0f / (float)nrows);
}